// SCAESuite_10316511445426
// MI455X (gfx1250) — compile-verified
//
#include <hip/hip_runtime.h>

typedef __attribute__((ext_vector_type(2))) float v2f;
typedef __attribute__((ext_vector_type(8))) float v8f;

#define NB     256     // batch
#define K_UP   64
#define K_DOWN 192
#define DD     768
#define DICT   24576
#define NC     256     // connections per dict entry
#define TILE_M 16      // kd rows per workgroup
#define KC     64      // K-chunk staged in LDS
#define NCHUNK (DD / KC)                // 12
#define TILES_PER_B (K_DOWN / TILE_M)   // 12
#define PAD    2

#if defined(__AMDGCN__) && __has_builtin(__builtin_amdgcn_global_load_async_to_lds_b32)
#define ASYNC_LDS 1
#else
#define ASYNC_LDS 0
#endif

#if ASYNC_LDS
typedef __attribute__((address_space(1))) int gint_t;   // global  (prints as __device__ int*)
typedef __attribute__((address_space(3))) int lint_t;   // LDS     (prints as __shared__ int*)

__device__ __forceinline__ void async_ld_b32(void* lds, const void* g) {
    // global addresses are numerically identical in generic and AS(1);
    // LDS needs a real addrspacecast (generic carries the shared aperture).
    __builtin_amdgcn_global_load_async_to_lds_b32(
        (gint_t*)(unsigned long long)(g), (lint_t*)lds, 0, 0);
}
__device__ __forceinline__ void wait_async0() {
#if __has_builtin(__builtin_amdgcn_s_wait_asynccnt)
    __builtin_amdgcn_s_wait_asynccnt(0);
#else
    asm volatile("s_wait_asynccnt 0" ::: "memory");
#endif
}
#endif

__global__ __launch_bounds__(128) void scae_pruned_contrib_kernel(
    const float* __restrict__ up_vals,      // [NB, K_UP]
    const float* __restrict__ up_decoder,   // [DD, DICT]
    const float* __restrict__ down_encoder, // [DICT, DD]
    const float* __restrict__ up_b_dec,     // [DD]
    const int*   __restrict__ up_indices,   // [NB, K_UP]
    const int*   __restrict__ down_indices, // [NB, K_DOWN]
    const int*   __restrict__ connections,  // [DICT, NC]
    float*       __restrict__ out)          // [NB, K_DOWN]
{
    const int b    = blockIdx.x / TILES_PER_B;
    const int kd0  = (blockIdx.x % TILES_PER_B) * TILE_M;
    const int tid  = threadIdx.x;      // 0..127
    const int lane = tid & 31;
    const int wave = tid >> 5;

    __shared__ int   up_idx_s[K_UP];
    __shared__ float up_val_s[K_UP];
    __shared__ int   dk_s[TILE_M];
    __shared__ float Asub[2][TILE_M][KC + PAD];   // double-buffered down_encoder chunk
    __shared__ float Bsub[2][KC][K_UP + PAD];     // double-buffered up_decoder gather
    __shared__ int   conn_s[TILE_M][NC];
    __shared__ float w_s[TILE_M][K_UP];
    __shared__ float part_match_s[TILE_M][8];
    __shared__ float part_bias_s[TILE_M][8];

    if (tid < K_UP) {
        up_idx_s[tid] = up_indices[b * K_UP + tid];
        up_val_s[tid] = up_vals[b * K_UP + tid];
    }
    if (tid < TILE_M) dk_s[tid] = down_indices[b * K_DOWN + kd0 + tid];
    __syncthreads();

    // ---- staging helper: issue one K-chunk into LDS buffer `buf` ----------
    auto stage = [&](int buf, int kb) {
        // A: 16 x 64 contiguous row pieces of down_encoder
        for (int i = tid; i < TILE_M * KC; i += 128) {
            const int m = i >> 6, k = i & (KC - 1);
            const float* g = down_encoder + (size_t)dk_s[m] * DD + kb + k;
#if ASYNC_LDS
            async_ld_b32(&Asub[buf][m][k], g);
#else
            Asub[buf][m][k] = *g;
#endif
        }
        // B: 64 x 64 scattered columns of up_decoder (L2-resident gather)
        for (int i = tid; i < KC * K_UP; i += 128) {
            const int k = i >> 6, u = i & (K_UP - 1);
            const float* g = up_decoder + (size_t)(kb + k) * DICT + up_idx_s[u];
#if ASYNC_LDS
            async_ld_b32(&Bsub[buf][k][u], g);
#else
            Bsub[buf][k][u] = *g;
#endif
        }
    };

    // kick off chunk 0 + connection rows (consumed only after the GEMM)
    stage(0, 0);
    for (int i = tid; i < TILE_M * NC; i += 128) {
        const int m = i >> 8, c = i & (NC - 1);
        const int* g = connections + (size_t)dk_s[m] * NC + c;
#if ASYNC_LDS
        async_ld_b32(&conn_s[m][c], g);
#else
        conn_s[m][c] = *g;
#endif
    }

    // ---- bias contribution: down_encoder[dk,:] . up_b_dec ----------------
    // (plain VMEM loads; overlap with the async staging above)
    {
        const int m = tid >> 3, seg = tid & 7;       // 16 rows x 8 segments of 96
        const float* __restrict__ row = down_encoder + (size_t)dk_s[m] * DD + seg * 96;
        const float* __restrict__ bv  = up_b_dec + seg * 96;
        float s = 0.f;
        #pragma unroll 8
        for (int i = 0; i < 96; ++i) s += row[i] * bv[i];
        part_bias_s[m][seg] = s;
    }

#if ASYNC_LDS
    wait_async0();
#endif
    __syncthreads();

    // ---- GEMM: vw tile via V_WMMA_F32_16X16X4_F32, double buffered --------
    v8f acc = {};
    const int mrow = lane & 15;            // M (A) / N (B) within 16x16 tile
    const int koff = (lane >> 4) * 2;      // lane-half selects K pair {0,1} vs {2,3}
    const int u0   = wave * 16;

    for (int kb = 0; kb < NCHUNK; ++kb) {
        const int cur = kb & 1;
        if (kb + 1 < NCHUNK) stage(cur ^ 1, (kb + 1) * KC);  // overlap next gather

        #pragma unroll
        for (int kk = 0; kk < KC; kk += 4) {
            v2f a, bf;
            a.x  = Asub[cur][mrow][kk + koff];
            a.y  = Asub[cur][mrow][kk + koff + 1];
            bf.x = Bsub[cur][kk + koff][u0 + mrow];
            bf.y = Bsub[cur][kk + koff + 1][u0 + mrow];
            acc = __builtin_amdgcn_wmma_f32_16x16x4_f32(
                false, a, false, bf, (short)0, acc, false, false);
        }
#if ASYNC_LDS
        wait_async0();
#endif
        __syncthreads();
    }

    // ---- scale by up_vals (per column u) and publish weighted tile --------
    {
        const int n = lane & 15;
        const int u = u0 + n;
        const float sc = up_val_s[u];
        #pragma unroll
        for (int j = 0; j < 8; ++j) {
            const int m = (lane < 16) ? j : j + 8;   // C/D layout: vgpr j -> M=j / M=j+8
            w_s[m][u] = acc[j] * sc;
        }
    }
    __syncthreads();

    // ---- match & accumulate: thread = (row m, 32-entry conn group) --------
    {
        int uix[K_UP];
        #pragma unroll
        for (int u = 0; u < K_UP; ++u) uix[u] = up_idx_s[u];

        const int m = tid >> 3, grp = tid & 7;
        float s = 0.f;
        for (int cc = 0; cc < 32; ++cc) {
            const int v = conn_s[m][grp * 32 + cc];
            if (v >= 0) {
                #pragma unroll
                for (int u = 0; u < K_UP; ++u)
                    if (uix[u] == v) s += w_s[m][u];   // duplicates counted, as in ref
            }
        }
        part_match_s[m][grp] = s;
    }
    __syncthreads();

    // ---- deterministic final reduction ------------------------------------
    if (tid < TILE_M) {
        float s = 0.f;
        #pragma unroll
        for (int g = 0; g < 8; ++g) s += part_match_s[tid][g] + part_bias_s[tid][g];
        out[b * K_DOWN + kd0 + tid] = s;
    }
}

extern "C" void kernel_launch(void* const* d_in, const int* in_sizes, int n_in,
                              void* d_out, int out_size, void* d_ws, size_t ws_size,
                              hipStream_t stream) {
    const float* up_vals      = (const float*)d_in[0];
    const float* up_decoder   = (const float*)d_in[1];
    const float* down_encoder = (const float*)d_in[2];
    const float* up_b_dec     = (const float*)d_in[3];
    const int*   up_indices   = (const int*)d_in[4];
    const int*   down_indices = (const int*)d_in[5];
    const int*   connections  = (const int*)d_in[6];
    float*       out          = (float*)d_out;

    const int grid = NB * TILES_PER_B;   // 3072 workgroups
    scae_pruned_contrib_kernel<<<grid, 128, 0, stream>>>(
        up_vals, up_decoder, down_encoder, up_b_dec,
        up_indices, down_indices, connections, out);
}